// GNN_37503654429443
// MI455X (gfx1250) — compile-verified
//
#include <hip/hip_runtime.h>
#include <hip/hip_bf16.h>

typedef __attribute__((ext_vector_type(2))) float v2f;
typedef __attribute__((ext_vector_type(8))) float v8f;

#define N_NODES   20000
#define N_EDGES   160000
#define HID       512
#define N_LAYERS  3
#define NUM_GRAPHS 128
#define BN_EPS    1e-5f

#define KT      64          // K-chunk staged in LDS
#define ROWPAD  68          // row pitch in LDS (floats): 68 % 64 == 4 -> conflict-free b64 frags
#define MTILE   128         // block M tile (8 waves x 16)
#define NTILE   64          // block N tile (4 wmma accumulators / wave)

// ---------------------------------------------------------------------------
// Edge message + scatter: aggr[dst] += ReLU(z[src] + ew)   (aggr pre-seeded with z)
// One thread = 4 consecutive features of one edge -> coalesced float4 traffic.
// ---------------------------------------------------------------------------
__global__ void edge_scatter_kernel(const float* __restrict__ z,
                                    const float* __restrict__ ew,
                                    const int*   __restrict__ ei,
                                    float*       __restrict__ aggr)
{
    long gid = (long)blockIdx.x * blockDim.x + threadIdx.x;
    int e  = (int)(gid >> 7);          // 128 threads per edge (512 feats / 4)
    int c  = (int)(gid & 127) << 2;
    if (e >= N_EDGES) return;
    int src = ei[e];
    int dst = ei[N_EDGES + e];
    const float4 a = *(const float4*)(z  + (size_t)src * HID + c);
    const float4 w = *(const float4*)(ew + (size_t)e   * HID + c);
    float4 m;
    m.x = fmaxf(a.x + w.x, 0.f);
    m.y = fmaxf(a.y + w.y, 0.f);
    m.z = fmaxf(a.z + w.z, 0.f);
    m.w = fmaxf(a.w + w.w, 0.f);
    float* d = aggr + (size_t)dst * HID + c;
    atomicAdd(d + 0, m.x);
    atomicAdd(d + 1, m.y);
    atomicAdd(d + 2, m.z);
    atomicAdd(d + 3, m.w);
}

// ---------------------------------------------------------------------------
// FP32 WMMA GEMM with LDS staging + register-prefetch double buffering.
//   out[m][n] = act( A'[m][:] . W[n][:] + bias[n] ),
//   A'[m][k] = BN_IN ? max(0, A[m][k]*scale[k] + shift[k]) : A[m][k]
// Block: 256 threads (8 waves), tile 128(M) x 64(N); K staged in 64-wide chunks.
// Fragment layout for V_WMMA_F32_16X16X4_F32 (wave32):
//   A frag: lane -> row (lane&15), v2f = A[m][kb], A[m][kb+1], kb = k0+2*(lane>>4)
//   B frag: same pattern over W rows (B[k][n] = W[n][k])
//   C/D:    lane&15 = N, vgpr v -> M = v + 8*(lane>>4)
// ---------------------------------------------------------------------------
template<bool BN_IN, bool RELU_OUT>
__global__ void gemm_wmma_kernel(const float* __restrict__ A,
                                 const float* __restrict__ W,
                                 const float* __restrict__ bias,
                                 const float* __restrict__ scale,
                                 const float* __restrict__ shift,
                                 float*       __restrict__ out,
                                 int nrows)
{
    __shared__ float sA[MTILE * ROWPAD];   // 128 x 64 (padded) = 34 KB
    __shared__ float sW[NTILE * ROWPAD];   //  64 x 64 (padded) = 17 KB

    const int tid  = threadIdx.x;
    const int wave = tid >> 5;             // 0..7
    const int lane = tid & 31;
    const int half = lane >> 4;            // 0 or 1
    const int l16  = lane & 15;
    const int m0   = blockIdx.x * MTILE + wave * 16;
    const int mblk = blockIdx.x * MTILE;
    const int n0   = blockIdx.y * NTILE;

    // Cooperative-load coordinates (constant per thread across chunks):
    // idx = tid + i*256 ; r = idx>>4 ; c4 = (idx&15)<<2  -> 256B coalesced rows.
    const int c4 = (tid & 15) << 2;

    float4 ra[8];                          // next A chunk (128x64 / 256 thr)
    float4 rw[4];                          // next W chunk ( 64x64 / 256 thr)
    float4 rs = {0, 0, 0, 0}, rt = {0, 0, 0, 0};

    auto prefetch = [&](int k0) {
        #pragma unroll
        for (int i = 0; i < 8; ++i) {
            int r = (tid + i * 256) >> 4;
            int row = mblk + r;
            if (row >= nrows) row = nrows - 1;       // clamp M edge (loads only)
            ra[i] = *(const float4*)(A + (size_t)row * HID + k0 + c4);
        }
        #pragma unroll
        for (int i = 0; i < 4; ++i) {
            int r = (tid + i * 256) >> 4;
            rw[i] = *(const float4*)(W + (size_t)(n0 + r) * HID + k0 + c4);
        }
        if (BN_IN) {
            rs = *(const float4*)(scale + k0 + c4);
            rt = *(const float4*)(shift + k0 + c4);
        }
    };

    v8f acc0 = {}, acc1 = {}, acc2 = {}, acc3 = {};

    prefetch(0);

    for (int k0 = 0; k0 < HID; k0 += KT) {
        __syncthreads();
        // Registers -> LDS (BN+ReLU applied once per A element here).
        #pragma unroll
        for (int i = 0; i < 8; ++i) {
            int r = (tid + i * 256) >> 4;
            float4 v = ra[i];
            if (BN_IN) {
                v.x = fmaxf(v.x * rs.x + rt.x, 0.f);
                v.y = fmaxf(v.y * rs.y + rt.y, 0.f);
                v.z = fmaxf(v.z * rs.z + rt.z, 0.f);
                v.w = fmaxf(v.w * rs.w + rt.w, 0.f);
            }
            *(float4*)(sA + r * ROWPAD + c4) = v;
        }
        #pragma unroll
        for (int i = 0; i < 4; ++i) {
            int r = (tid + i * 256) >> 4;
            *(float4*)(sW + r * ROWPAD + c4) = rw[i];
        }
        __syncthreads();

        // Issue next chunk's global loads; they fly while we do 64 WMMAs.
        if (k0 + KT < HID) prefetch(k0 + KT);

        const float* pa = sA + (wave * 16 + l16) * ROWPAD + 2 * half;
        const float* p0 = sW + ( 0 + l16) * ROWPAD + 2 * half;
        const float* p1 = sW + (16 + l16) * ROWPAD + 2 * half;
        const float* p2 = sW + (32 + l16) * ROWPAD + 2 * half;
        const float* p3 = sW + (48 + l16) * ROWPAD + 2 * half;

        #pragma unroll
        for (int kt = 0; kt < KT; kt += 4) {
            v2f a  = *(const v2f*)(pa + kt);
            v2f b0 = *(const v2f*)(p0 + kt);
            v2f b1 = *(const v2f*)(p1 + kt);
            v2f b2 = *(const v2f*)(p2 + kt);
            v2f b3 = *(const v2f*)(p3 + kt);
            acc0 = __builtin_amdgcn_wmma_f32_16x16x4_f32(false, a, false, b0, (short)0, acc0, false, false);
            acc1 = __builtin_amdgcn_wmma_f32_16x16x4_f32(false, a, false, b1, (short)0, acc1, false, false);
            acc2 = __builtin_amdgcn_wmma_f32_16x16x4_f32(false, a, false, b2, (short)0, acc2, false, false);
            acc3 = __builtin_amdgcn_wmma_f32_16x16x4_f32(false, a, false, b3, (short)0, acc3, false, false);
        }
    }

    // Epilogue: bias (+ optional ReLU), guarded stores on the M edge.
    #pragma unroll
    for (int v = 0; v < 8; ++v) {
        const int row = m0 + v + half * 8;
        if (row < nrows) {
            {
                const int col = n0 + 0 + l16;
                float val = acc0[v] + bias[col];
                if (RELU_OUT) val = fmaxf(val, 0.f);
                out[(size_t)row * HID + col] = val;
            }
            {
                const int col = n0 + 16 + l16;
                float val = acc1[v] + bias[col];
                if (RELU_OUT) val = fmaxf(val, 0.f);
                out[(size_t)row * HID + col] = val;
            }
            {
                const int col = n0 + 32 + l16;
                float val = acc2[v] + bias[col];
                if (RELU_OUT) val = fmaxf(val, 0.f);
                out[(size_t)row * HID + col] = val;
            }
            {
                const int col = n0 + 48 + l16;
                float val = acc3[v] + bias[col];
                if (RELU_OUT) val = fmaxf(val, 0.f);
                out[(size_t)row * HID + col] = val;
            }
        }
    }
}

// ---------------------------------------------------------------------------
// BatchNorm stats: per-column sum / sum-of-squares over nrows (coalesced:
// 512 threads = 512 columns, block strides a row chunk, atomics to combine).
// ---------------------------------------------------------------------------
__global__ void bn_stats_kernel(const float* __restrict__ h,
                                float* __restrict__ sum,
                                float* __restrict__ sumsq,
                                int nrows, int rows_per_block)
{
    const int col = threadIdx.x;             // blockDim.x == 512
    int r0 = blockIdx.x * rows_per_block;
    int r1 = r0 + rows_per_block;
    if (r1 > nrows) r1 = nrows;
    float s = 0.f, ss = 0.f;
    for (int r = r0; r < r1; ++r) {
        float v = h[(size_t)r * HID + col];
        s  += v;
        ss += v * v;
    }
    atomicAdd(&sum[col], s);
    atomicAdd(&sumsq[col], ss);
}

__global__ void bn_finalize_kernel(const float* __restrict__ sum,
                                   const float* __restrict__ sumsq,
                                   const float* __restrict__ gamma,
                                   const float* __restrict__ beta,
                                   float* __restrict__ scale,
                                   float* __restrict__ shift,
                                   int nrows)
{
    const int c = threadIdx.x;               // blockDim.x == 512
    const float inv_n = 1.f / (float)nrows;
    float mu   = sum[c] * inv_n;
    float var  = sumsq[c] * inv_n - mu * mu;
    float rstd = rsqrtf(var + BN_EPS);
    float sc   = gamma[c] * rstd;
    scale[c] = sc;
    shift[c] = beta[c] - mu * sc;
}

// ---------------------------------------------------------------------------
// Graph readout: g[batch[n]][layer*HID + c] += z[n][c]
// ---------------------------------------------------------------------------
__global__ void readout_kernel(const float* __restrict__ z,
                               const int*   __restrict__ batch,
                               float*       __restrict__ g,
                               int layer)
{
    long gid = (long)blockIdx.x * blockDim.x + threadIdx.x;
    int node = (int)(gid >> 7);
    int c    = (int)(gid & 127) << 2;
    if (node >= N_NODES) return;
    int b = batch[node];
    const float4 v = *(const float4*)(z + (size_t)node * HID + c);
    float* d = g + (size_t)b * (N_LAYERS * HID) + (size_t)layer * HID + c;
    atomicAdd(d + 0, v.x);
    atomicAdd(d + 1, v.y);
    atomicAdd(d + 2, v.z);
    atomicAdd(d + 3, v.w);
}

// ---------------------------------------------------------------------------
extern "C" void kernel_launch(void* const* d_in, const int* in_sizes, int n_in,
                              void* d_out, int out_size, void* d_ws, size_t ws_size,
                              hipStream_t stream)
{
    const float* x     = (const float*)d_in[0];
    const int*   ei    = (const int*)  d_in[1];
    const float* ew    = (const float*)d_in[2];
    const int*   batch = (const int*)  d_in[3];
    const float* w1    = (const float*)d_in[4];
    const float* b1    = (const float*)d_in[5];
    const float* gamma = (const float*)d_in[6];
    const float* beta  = (const float*)d_in[7];
    const float* w2    = (const float*)d_in[8];
    const float* b2    = (const float*)d_in[9];

    const size_t NH = (size_t)N_NODES * HID;
    float* B0    = (float*)d_ws;              // N x H
    float* B1    = B0 + NH;                   // N x H
    float* stats = B1 + NH;                   // sum[512], sumsq[512], scale[512], shift[512]
    float* s_sum   = stats;
    float* s_sumsq = stats + HID;
    float* s_scale = stats + 2 * HID;
    float* s_shift = stats + 3 * HID;

    float* out_z = (float*)d_out;             // N x H  (final layer z)
    float* out_g = out_z + NH;                // G x (L*H)

    // g region must be zeroed every call (atomically accumulated).
    hipMemsetAsync(out_g, 0, (size_t)NUM_GRAPHS * N_LAYERS * HID * sizeof(float), stream);

    // Buffer rotation (3 node-feature buffers: B0, B1, out_z):
    //   L0: zcur=x,     aggr=B0, h=B1, zout=out_z
    //   L1: zcur=out_z, aggr=B0, h=B1, zout=B0   (aggr dead after GEMM1)
    //   L2: zcur=B0,    aggr=B1, h=B0, zout=out_z
    const float* zcur_tab[N_LAYERS] = { x,     out_z, B0    };
    float*       aggr_tab[N_LAYERS] = { B0,    B0,    B1    };
    float*       h_tab   [N_LAYERS] = { B1,    B1,    B0    };
    float*       zout_tab[N_LAYERS] = { out_z, B0,    out_z };

    const dim3 gemm_grid((N_NODES + MTILE - 1) / MTILE, HID / NTILE);
    const dim3 gemm_block(256);

    const int edge_blocks = (int)(((long)N_EDGES * 128 + 255) / 256);
    const int node_blocks = (int)(((long)N_NODES * 128 + 255) / 256);
    const int bn_blocks = 128;
    const int rows_per_block = (N_NODES + bn_blocks - 1) / bn_blocks;

    for (int l = 0; l < N_LAYERS; ++l) {
        const float* zcur = zcur_tab[l];
        float* aggr = aggr_tab[l];
        float* h    = h_tab[l];
        float* zout = zout_tab[l];

        // aggr = zcur, then aggr[dst] += ReLU(zcur[src] + ew)
        hipMemcpyAsync(aggr, zcur, NH * sizeof(float), hipMemcpyDeviceToDevice, stream);
        edge_scatter_kernel<<<edge_blocks, 256, 0, stream>>>(zcur, ew, ei, aggr);

        // h = aggr @ w1[l]^T + b1[l]
        gemm_wmma_kernel<false, false><<<gemm_grid, gemm_block, 0, stream>>>(
            aggr, w1 + (size_t)l * HID * HID, b1 + (size_t)l * HID,
            nullptr, nullptr, h, N_NODES);

        // BatchNorm stats -> fused affine coefficients
        hipMemsetAsync(s_sum, 0, 2 * HID * sizeof(float), stream);
        bn_stats_kernel<<<bn_blocks, HID, 0, stream>>>(h, s_sum, s_sumsq, N_NODES, rows_per_block);
        bn_finalize_kernel<<<1, HID, 0, stream>>>(s_sum, s_sumsq,
            gamma + (size_t)l * HID, beta + (size_t)l * HID, s_scale, s_shift, N_NODES);

        // z = ReLU( ReLU(BN(h)) @ w2[l]^T + b2[l] )   (BN+ReLU fused into A staging)
        gemm_wmma_kernel<true, true><<<gemm_grid, gemm_block, 0, stream>>>(
            h, w2 + (size_t)l * HID * HID, b2 + (size_t)l * HID,
            s_scale, s_shift, zout, N_NODES);

        // g[:, l*H : (l+1)*H] = segment_sum(z, batch)
        readout_kernel<<<node_blocks, 256, 0, stream>>>(zout, batch, out_g, l);
    }
}